// EnhancedDiffusionLayer_31009663877298
// MI455X (gfx1250) — compile-verified
//
#include <hip/hip_runtime.h>
#include <hip/hip_bf16.h>

// Problem constants (match reference)
#define BB 16
#define CC 8
#define SS 128
#define NSTEPS 10
#define EPSV 1e-6f

typedef __attribute__((ext_vector_type(2))) float v2f;
typedef __attribute__((ext_vector_type(8))) float v8f;

// Fast hardware reciprocal (v_rcp_f32, <=1 ulp). denom here is ~1+2*coeff
// (diagonally dominant), so this is numerically safe and removes the long
// IEEE-div sequence from the serial Thomas recurrence (the critical path).
__device__ __forceinline__ float fast_rcp(float x) {
  return __builtin_amdgcn_rcpf(x);
}

// ---------------------------------------------------------------------------
// Fused channel-mix (optional) + implicit x-direction (along W) Thomas solve.
// One block per (b, h). Block = 256 threads = 8 waves.
// LDS tile T[w][c] with padded stride 9 (bank-conflict free), coefficients
// precomputed into CO so the serial Thomas recurrence never touches global.
// Channel mixing (out[d] = sum_c M[d,c] * u[c]) is done with two chained
// V_WMMA_F32_16X16X4_F32 ops per wave (K=0..3 then K=4..7), 16 spatial
// points per wave-tile, 8 waves covering all 128 points of the row.
// ---------------------------------------------------------------------------
template <bool DO_MIX>
__global__ __launch_bounds__(256) void diffuse_x_kernel(
    const float* __restrict__ src, float* __restrict__ dst,
    const float* __restrict__ alpha_base, const float* __restrict__ alpha_tc,
    const float* __restrict__ mixing, float tcur, float scale) {
  __shared__ float Tls[SS * 9];     // tile: [w][c], stride 9
  __shared__ float CO[SS * 9];      // coeff: [w][c], stride 9
  __shared__ float Mls[CC * CC];    // mixing matrix, row-major [d][c]
  __shared__ float CSs[SS * CC];    // c_star: [i][c]
  __shared__ float DSs[SS * CC];    // d_star: [i][c]

  const int tid = threadIdx.x;
  const int b = blockIdx.x >> 7;    // /128
  const int h = blockIdx.x & 127;

  if (DO_MIX) {
    if (tid < CC * CC) Mls[tid] = mixing[tid];
  }

  // Phase 1: coalesced load of u[b,:,h,:] and coefficient precompute.
  for (int k = 0; k < 4; ++k) {
    const int e = k * 256 + tid;            // 0..1023
    const int c = e >> 7;                   // 0..7
    const int w = e & 127;
    const int gsp = ((b * CC + c) * SS + h) * SS + w;
    const int gcf = (c * SS + h) * SS + w;
    Tls[w * 9 + c] = src[gsp];
    const float co =
        fminf(fmaxf(fmaf(alpha_tc[gcf], tcur, alpha_base[gcf]), EPSV), 10.0f) *
        scale;
    CO[w * 9 + c] = co;
  }
  __syncthreads();

  // Phase 2: channel mixing via f32 WMMA (in-place in LDS, per-wave tiles).
  if (DO_MIX) {
    const int lane = tid & 31;
    const int wv = tid >> 5;                // wave id 0..7
    const int w0 = wv * 16;                 // this wave's 16 spatial points
    const int r = lane & 15;
    const int hi = (lane >= 16) ? 1 : 0;
    const int ka = hi * 2;                  // K offset for this lane half

    // A fragment: A[m][k] = M[m][k] for m<8, zero-padded rows 8..15.
    v2f a0, a1;
    {
      const int rr = r & 7;
      const float z = (r < 8) ? 1.0f : 0.0f;
      a0.x = z * Mls[rr * 8 + ka];
      a0.y = z * Mls[rr * 8 + ka + 1];
      a1.x = z * Mls[rr * 8 + 4 + ka];
      a1.y = z * Mls[rr * 8 + 4 + ka + 1];
    }
    // B fragment: B[k][n] = u[channel=k][w0+n].
    const float* col = &Tls[(w0 + r) * 9];
    v2f b0, b1;
    b0.x = col[ka];
    b0.y = col[ka + 1];
    b1.x = col[4 + ka];
    b1.y = col[4 + ka + 1];

    v8f acc = {0.f, 0.f, 0.f, 0.f, 0.f, 0.f, 0.f, 0.f};
    acc = __builtin_amdgcn_wmma_f32_16x16x4_f32(false, a0, false, b0,
                                                (short)0, acc, false, false);
    acc = __builtin_amdgcn_wmma_f32_16x16x4_f32(false, a1, false, b1,
                                                (short)0, acc, false, false);

    // D rows 0..7 (lanes 0..15) are the 8 mixed channels at point w0+r.
    if (!hi) {
      float* colo = &Tls[(w0 + r) * 9];
#pragma unroll
      for (int d = 0; d < CC; ++d) colo[d] = acc[d];
    }
    __syncthreads();
  }

  // Phase 3: serial Thomas solve per channel row (8 active threads),
  // entirely in LDS, with fast-reciprocal on the recurrence.
  if (tid < CC) {
    const int c = tid;
    float cs_p = 0.f, ds_p = 0.f;
#pragma unroll 4
    for (int i = 0; i < SS; ++i) {
      const float co = CO[i * 9 + c];
      const float a = -co;
      const float bco = (i == 0 || i == SS - 1) ? (1.0f + co) : (1.0f + 2.0f * co);
      const float d = Tls[i * 9 + c];
      const float denom = bco - a * cs_p + EPSV;
      const float inv = fast_rcp(denom);
      const float cs = (-co) * inv;
      const float ds = (d - a * ds_p) * inv;
      CSs[i * CC + c] = cs;
      DSs[i * CC + c] = ds;
      cs_p = cs;
      ds_p = ds;
    }
    float x = ds_p;                          // x[N-1] = d_star[N-1]
    Tls[(SS - 1) * 9 + c] = x;
#pragma unroll 4
    for (int i = SS - 2; i >= 0; --i) {
      x = DSs[i * CC + c] - CSs[i * CC + c] * x;
      Tls[i * 9 + c] = x;
    }
  }
  __syncthreads();

  // Phase 4: coalesced store back.
  for (int k = 0; k < 4; ++k) {
    const int e = k * 256 + tid;
    const int c = e >> 7;
    const int w = e & 127;
    dst[((b * CC + c) * SS + h) * SS + w] = Tls[w * 9 + c];
  }
}

// ---------------------------------------------------------------------------
// Implicit y-direction (along H) Thomas solve, in-place on u.
// One 64-thread block handles 64 consecutive w-columns of one (b,c) image:
// all global accesses are lane-coalesced. c*/d* live in 64KB LDS; next-row
// loads are manually prefetched to overlap with the recurrence chain.
// ---------------------------------------------------------------------------
__global__ __launch_bounds__(64) void diffuse_y_kernel(
    float* __restrict__ u, const float* __restrict__ beta_base,
    const float* __restrict__ beta_tc, float tcur, float scale) {
  __shared__ float CSs[SS * 64];
  __shared__ float DSs[SS * 64];

  const int tid = threadIdx.x;
  const int bc = blockIdx.x >> 1;                 // (b*C + c)
  const int w = ((blockIdx.x & 1) << 6) + tid;    // column index
  const int c = bc & (CC - 1);

  float* up = u + bc * SS * SS + w;               // element i at up[i*SS]
  const float* bbp = beta_base + c * SS * SS + w;
  const float* btp = beta_tc + c * SS * SS + w;

  float cs_p = 0.f, ds_p = 0.f;
  float d_n = up[0], ab_n = bbp[0], at_n = btp[0];
  for (int i = 0; i < SS; ++i) {
    const float d = d_n, ab = ab_n, at = at_n;
    if (i < SS - 1) {
      d_n = up[(i + 1) * SS];
      ab_n = bbp[(i + 1) * SS];
      at_n = btp[(i + 1) * SS];
    }
    const float co = fminf(fmaxf(fmaf(at, tcur, ab), EPSV), 10.0f) * scale;
    const float a = -co;
    const float bco = (i == 0 || i == SS - 1) ? (1.0f + co) : (1.0f + 2.0f * co);
    const float denom = bco - a * cs_p + EPSV;
    const float inv = fast_rcp(denom);
    const float cs = (-co) * inv;
    const float ds = (d - a * ds_p) * inv;
    CSs[i * 64 + tid] = cs;
    DSs[i * 64 + tid] = ds;
    cs_p = cs;
    ds_p = ds;
  }
  float x = ds_p;
  up[(SS - 1) * SS] = x;
  for (int i = SS - 2; i >= 0; --i) {
    x = DSs[i * 64 + tid] - CSs[i * 64 + tid] * x;
    up[i * SS] = x;
  }
}

// ---------------------------------------------------------------------------
// Host driver: 10 ADI steps, 3 launches each. In-place in d_out after the
// first mix+x pass consumes the (immutable) input. Deterministic; no d_ws use.
// ---------------------------------------------------------------------------
extern "C" void kernel_launch(void* const* d_in, const int* in_sizes, int n_in,
                              void* d_out, int out_size, void* d_ws,
                              size_t ws_size, hipStream_t stream) {
  (void)in_sizes; (void)n_in; (void)out_size; (void)d_ws; (void)ws_size;

  const float* u_in = (const float*)d_in[0];
  const float* alpha_base = (const float*)d_in[1];
  const float* beta_base = (const float*)d_in[2];
  const float* alpha_tc = (const float*)d_in[3];
  const float* beta_tc = (const float*)d_in[4];
  const float* mixing = (const float*)d_in[5];
  float* u = (float*)d_out;

  const double DT = 0.001, HALF = 0.0005;
  const float sx = (float)HALF;  // half / DX^2
  const float sy = (float)DT;    // DT / DY^2

  const dim3 gx(BB * SS), bx(256);
  const dim3 gy(BB * CC * 2), by(64);

  for (int s = 0; s < NSTEPS; ++s) {
    const float t0 = (float)(s * DT);
    const float t1 = (float)(s * DT + HALF);
    const float t2 = (float)((s + 1) * DT);
    const float* src = (s == 0) ? u_in : (const float*)u;

    diffuse_x_kernel<true><<<gx, bx, 0, stream>>>(src, u, alpha_base, alpha_tc,
                                                  mixing, t0, sx);
    diffuse_y_kernel<<<gy, by, 0, stream>>>(u, beta_base, beta_tc, t1, sy);
    diffuse_x_kernel<false><<<gx, bx, 0, stream>>>(u, u, alpha_base, alpha_tc,
                                                   mixing, t2, sx);
  }
}